// LSTMAttentionDot_68478958567476
// MI455X (gfx1250) — compile-verified
//
#include <hip/hip_runtime.h>
#include <math.h>

// Problem dims (fixed by reference)
#define BB   32
#define TT   256
#define SS   128
#define INW  512
#define HH   512
#define H2   1024   // 2*H
#define G4H  2048   // 4*H

#define ASTR512  516   // padded LDS row stride for K=512  (bank-conflict free)
#define ASTR1024 1028  // padded LDS row stride for K=1024

typedef float v2f __attribute__((ext_vector_type(2)));
typedef float v8f __attribute__((ext_vector_type(8)));

// D = A(16x4 f32) * B(4x16 f32) + C(16x16 f32), wave32
__device__ __forceinline__ v8f wmma_f32_k4(v2f a, v2f b, v8f c) {
    return __builtin_amdgcn_wmma_f32_16x16x4_f32(
        /*neg_a=*/false, a, /*neg_b=*/false, b,
        /*c_mod=*/(short)0, c, /*reuse_a=*/false, /*reuse_b=*/false);
}

// WMMA K-loop: A fragment from LDS slab (16 rows, padded stride ASTR),
// B fragment from K-major global matrix BT with leading dim LDB.
template <int KLEN, int LDB, int ASTR>
__device__ __forceinline__ v8f wmma_kloop_lds(const float* __restrict__ As,
                                              const float* __restrict__ BT,
                                              int n0, int lane, v8f acc) {
    const int lrow  = lane & 15;
    const int khalf = (lane >> 4) << 1;
    const float* __restrict__ ap  = As + lrow * ASTR + khalf;
    const float* __restrict__ bp0 = BT + (size_t)khalf * LDB + (n0 + lrow);
    for (int k = 0; k < KLEN; k += 4) {
        v2f a, b;
        a.x = ap[k];
        a.y = ap[k + 1];
        const float* bp = bp0 + (size_t)k * LDB;
        b.x = bp[0];
        b.y = bp[LDB];
        acc = wmma_f32_k4(a, b, acc);
    }
    return acc;
}

// -------------------------------------------------------------------------
// 32x32 LDS tile transpose: src[R][C] -> dst[C][R]  (R, C multiples of 32)
// -------------------------------------------------------------------------
__global__ __launch_bounds__(256) void transpose_kernel(
    const float* __restrict__ src, float* __restrict__ dst, int R, int C) {
    __shared__ float tile[32][33];
    const int bc = blockIdx.x * 32;
    const int br = blockIdx.y * 32;
    const int tx = threadIdx.x;
    const int ty = threadIdx.y;
    for (int i = ty; i < 32; i += 8)
        tile[i][tx] = src[(size_t)(br + i) * C + bc + tx];
    __syncthreads();
    for (int i = ty; i < 32; i += 8)
        dst[(size_t)(bc + i) * R + br + tx] = tile[tx][i];
}

// -------------------------------------------------------------------------
// x_gates[t*B+b][n] = sum_k x[b][t][k] * WiT[k][n] + (bi[n]+bh[n])
// Block = 8 waves sharing one 16-row A slab (LDS). M=8192, N=2048, K=512.
// mt = blockIdx/16 (0..511), nt = (blockIdx%16)*8 + waveInBlock (0..127).
// -------------------------------------------------------------------------
__global__ __launch_bounds__(256) void xgates_kernel(
    const float* __restrict__ x, const float* __restrict__ WiT,
    const float* __restrict__ bi, const float* __restrict__ bh,
    float* __restrict__ xg) {
    __shared__ float As[16 * ASTR512];
    const int tid  = threadIdx.x;
    const int lane = tid & 31;
    const int widx = tid >> 5;                    // 0..7
    const int mt   = (int)blockIdx.x >> 4;        // 0..511
    const int nt   = (((int)blockIdx.x & 15) << 3) + widx;
    const int m0   = mt * 16;
    const int n0   = nt * 16;

    // Cooperative A-slab copy: rows m0..m0+15; m = t*B+b -> b=m&31, t=m>>5.
    for (int i = tid; i < 16 * INW; i += 256) {
        const int r = i >> 9;                     // row in slab
        const int c = i & (INW - 1);
        const int m = m0 + r;
        As[r * ASTR512 + c] =
            x[(size_t)(m & 31) * TT * INW + (size_t)(m >> 5) * INW + c];
    }
    __syncthreads();

    v8f acc = {};
    acc = wmma_kloop_lds<INW, G4H, ASTR512>(As, WiT, n0, lane, acc);

    const int col   = n0 + (lane & 15);
    const float add = bi[col] + bh[col];
    const int rbase = m0 + ((lane >> 4) << 3);
#pragma unroll
    for (int v = 0; v < 8; ++v)
        xg[(size_t)(rbase + v) * G4H + col] = acc[v] + add;
}

// -------------------------------------------------------------------------
// ctxP = ctx @ W_in : M=4096 (ctx rows contiguous), N=512, K=512.
// W_in [H][H] row-major is already K-major for the B fragment.
// mt = blockIdx/4 (0..255), nt = (blockIdx%4)*8 + waveInBlock (0..31).
// -------------------------------------------------------------------------
__global__ __launch_bounds__(256) void ctxp_kernel(
    const float* __restrict__ ctx, const float* __restrict__ W_in,
    float* __restrict__ ctxP) {
    __shared__ float As[16 * ASTR512];
    const int tid  = threadIdx.x;
    const int lane = tid & 31;
    const int widx = tid >> 5;
    const int mt   = (int)blockIdx.x >> 2;
    const int nt   = (((int)blockIdx.x & 3) << 3) + widx;
    const int m0   = mt * 16;
    const int n0   = nt * 16;

    const float* __restrict__ slab = ctx + (size_t)m0 * HH;  // contiguous
    for (int i = tid; i < 16 * HH; i += 256)
        As[(i >> 9) * ASTR512 + (i & (HH - 1))] = slab[i];
    __syncthreads();

    v8f acc = {};
    acc = wmma_kloop_lds<HH, HH, ASTR512>(As, W_in, n0, lane, acc);

    const int col   = n0 + (lane & 15);
    const int rbase = m0 + ((lane >> 4) << 3);
#pragma unroll
    for (int v = 0; v < 8; ++v)
        ctxP[(size_t)(rbase + v) * HH + col] = acc[v];
}

// -------------------------------------------------------------------------
// Per-step gate GEMM:
//   gates[b][n] = sum_k h[b][k]*WhT[k][n] + (precomp ? xg_t[b][n]
//                                                    : x_t row GEMM + bi+bh)
// 32 blocks x 8 waves; mt = blockIdx/16 (0..1), nt = (blockIdx%16)*8 + wave.
// h slab (16 rows) staged in LDS once per block.
// -------------------------------------------------------------------------
__global__ __launch_bounds__(256) void step_gates_kernel(
    const float* __restrict__ h, const float* __restrict__ WhT,
    const float* __restrict__ xg_t,   // [B][4H] slice (precomp mode) or null
    const float* __restrict__ x_t,    // x + t*INW, row stride T*INW (fold mode)
    const float* __restrict__ WiT,
    const float* __restrict__ bi, const float* __restrict__ bh,
    float* __restrict__ gates, int fold) {
    __shared__ float As[16 * ASTR512];
    const int tid  = threadIdx.x;
    const int lane = tid & 31;
    const int widx = tid >> 5;
    const int mt   = (int)blockIdx.x >> 4;        // 0..1
    const int nt   = (((int)blockIdx.x & 15) << 3) + widx;
    const int m0   = mt * 16;
    const int n0   = nt * 16;

    const float* __restrict__ slab = h + (size_t)m0 * HH;    // contiguous
    for (int i = tid; i < 16 * HH; i += 256)
        As[(i >> 9) * ASTR512 + (i & (HH - 1))] = slab[i];
    __syncthreads();

    v8f acc = {};
    acc = wmma_kloop_lds<HH, G4H, ASTR512>(As, WhT, n0, lane, acc);

    const int lrow  = lane & 15;
    const int khalf = (lane >> 4) << 1;
    if (fold) {
        // x_t (32x512, row stride T*INW) * WiT (512x2048), global A path
        const float* __restrict__ arow = x_t + (size_t)(m0 + lrow) * TT * INW;
        for (int k = 0; k < INW; k += 4) {
            v2f a, b;
            a.x = arow[k + khalf];
            a.y = arow[k + khalf + 1];
            const float* bp = WiT + (size_t)(k + khalf) * G4H + (n0 + lrow);
            b.x = bp[0];
            b.y = bp[G4H];
            acc = wmma_f32_k4(a, b, acc);
        }
    }

    const int col   = n0 + lrow;
    const int rbase = m0 + ((lane >> 4) << 3);
    if (fold) {
        const float add = bi[col] + bh[col];
#pragma unroll
        for (int v = 0; v < 8; ++v)
            gates[(size_t)(rbase + v) * G4H + col] = acc[v] + add;
    } else {
#pragma unroll
        for (int v = 0; v < 8; ++v)
            gates[(size_t)(rbase + v) * G4H + col] =
                acc[v] + xg_t[(size_t)(rbase + v) * G4H + col];
    }
}

// -------------------------------------------------------------------------
// Per-step per-batch kernel: LSTM cell -> hy; scores via precomputed ctxP;
// softmax; weighted sum. Only b-local data (no weight matrices).
// Writes hy into cath[b][H..2H) and weighted into cath[b][0..H).
// -------------------------------------------------------------------------
__device__ __forceinline__ float wave_sum32(float s) {
#pragma unroll
    for (int off = 16; off > 0; off >>= 1) s += __shfl_xor(s, off, 32);
    return s;
}
__device__ __forceinline__ float wave_max32(float s) {
#pragma unroll
    for (int off = 16; off > 0; off >>= 1) s = fmaxf(s, __shfl_xor(s, off, 32));
    return s;
}

__global__ __launch_bounds__(512) void step_attn_kernel(
    const float* __restrict__ gates,           // [B][4H]
    float* __restrict__ cbuf,                  // [B][H] (in/out)
    const float* __restrict__ ctx,             // [B][S][H]
    const float* __restrict__ ctxP,            // [B][S][H] = ctx @ W_in
    const unsigned char* __restrict__ mask,    // [B][S] bool
    float* __restrict__ cath) {                // [B][2H]: [w | hy]
    __shared__ float hy_s[HH];
    __shared__ float sc_s[SS];
    __shared__ float red_s;

    const int b    = blockIdx.x;
    const int tid  = threadIdx.x;
    const int lane = tid & 31;
    const int wid  = tid >> 5;   // 0..15

    // ---- LSTM cell ----
    {
        const float* g = gates + (size_t)b * G4H;
        const int k = tid;
        const float ig = g[k];
        const float fg = g[k + HH];
        const float gg = g[k + 2 * HH];
        const float og = g[k + 3 * HH];
        const float si = 1.0f / (1.0f + __expf(-ig));
        const float sf = 1.0f / (1.0f + __expf(-fg));
        const float so = 1.0f / (1.0f + __expf(-og));
        const float cn = sf * cbuf[(size_t)b * HH + k] + si * tanhf(gg);
        cbuf[(size_t)b * HH + k] = cn;
        const float hyv = so * tanhf(cn);
        hy_s[k] = hyv;
        cath[(size_t)b * H2 + HH + k] = hyv;
    }
    __syncthreads();

    // ---- scores[s] = hy . ctxP[b,s,:] + mask_add ----
    const float* cP = ctxP + (size_t)b * SS * HH;
    for (int si = wid; si < SS; si += 16) {
        const float* cr = cP + (size_t)si * HH;
        __builtin_prefetch(cP + (size_t)(si + 16) * HH + lane * 16, 0, 0);
        float s = 0.f;
        for (int k = lane; k < HH; k += 32) s += cr[k] * hy_s[k];
        s = wave_sum32(s);
        if (lane == 0) {
            const float madd = mask[(size_t)b * SS + si] ? 0.0f : -1.0e10f;
            sc_s[si] = s + madd;
        }
    }
    __syncthreads();

    // ---- softmax over S=128 (wave 0) ----
    if (wid == 0) {
        float m = -3.4e38f;
        for (int i = lane; i < SS; i += 32) m = fmaxf(m, sc_s[i]);
        m = wave_max32(m);
        float sum = 0.f;
        for (int i = lane; i < SS; i += 32) {
            const float e = __expf(sc_s[i] - m);
            sc_s[i] = e;
            sum += e;
        }
        sum = wave_sum32(sum);
        if (lane == 0) red_s = 1.0f / sum;
    }
    __syncthreads();
    const float inv = red_s;

    // ---- weighted[k] = sum_s alpha[s] * ctx[b,s,k] ----
    {
        const float* cb = ctx + (size_t)b * SS * HH;
        const int k = tid;
        float s = 0.f;
        for (int si = 0; si < SS; ++si) s += sc_s[si] * cb[(size_t)si * HH + k];
        cath[(size_t)b * H2 + k] = s * inv;
    }
}

// -------------------------------------------------------------------------
// h_tilde = tanh(cath(32x1024) @ WoT(1024x512)), WMMA, A slab in LDS.
// 8 blocks; mt = blockIdx/4 (0..1), nt = (blockIdx%4)*8 + wave (0..31).
// -------------------------------------------------------------------------
__global__ __launch_bounds__(256) void step_htilde_kernel(
    const float* __restrict__ cath, const float* __restrict__ WoT,
    float* __restrict__ hbuf, float* __restrict__ out_t) {
    __shared__ float As[16 * ASTR1024];
    const int tid  = threadIdx.x;
    const int lane = tid & 31;
    const int widx = tid >> 5;
    const int mt   = (int)blockIdx.x >> 2;        // 0..1
    const int nt   = (((int)blockIdx.x & 3) << 3) + widx;
    const int m0   = mt * 16;
    const int n0   = nt * 16;

    const float* __restrict__ slab = cath + (size_t)m0 * H2;  // contiguous
    for (int i = tid; i < 16 * H2; i += 256)
        As[(i >> 10) * ASTR1024 + (i & (H2 - 1))] = slab[i];
    __syncthreads();

    v8f acc = {};
    acc = wmma_kloop_lds<H2, HH, ASTR1024>(As, WoT, n0, lane, acc);

    const int col   = n0 + (lane & 15);
    const int rbase = m0 + ((lane >> 4) << 3);
#pragma unroll
    for (int v = 0; v < 8; ++v) {
        const int row = rbase + v;               // batch index b
        const float ht = tanhf(acc[v]);
        hbuf[(size_t)row * HH + col] = ht;
        out_t[(size_t)row * TT * HH + col] = ht;
    }
}

// -------------------------------------------------------------------------
extern "C" void kernel_launch(void* const* d_in, const int* in_sizes, int n_in,
                              void* d_out, int out_size, void* d_ws, size_t ws_size,
                              hipStream_t stream) {
    const float*         x     = (const float*)d_in[0];
    const float*         h0    = (const float*)d_in[1];
    const float*         c0    = (const float*)d_in[2];
    const float*         ctx   = (const float*)d_in[3];
    const unsigned char* cmask = (const unsigned char*)d_in[4];
    const float*         Wi    = (const float*)d_in[5];
    const float*         bi    = (const float*)d_in[6];
    const float*         Wh    = (const float*)d_in[7];
    const float*         bh    = (const float*)d_in[8];
    const float*         W_in  = (const float*)d_in[9];
    const float*         W_out = (const float*)d_in[10];
    float*               out   = (float*)d_out;

    // Workspace layout (floats)
    float* ws = (float*)d_ws;
    size_t off = 0;
    float* WiT   = ws + off; off += (size_t)INW * G4H;     // 4 MB
    float* WhT   = ws + off; off += (size_t)HH  * G4H;     // 4 MB
    float* WoT   = ws + off; off += (size_t)H2  * HH;      // 2 MB
    float* ctxP  = ws + off; off += (size_t)BB * SS * HH;  // 8 MB
    float* hbuf  = ws + off; off += (size_t)BB * HH;
    float* cbuf  = ws + off; off += (size_t)BB * HH;
    float* gates = ws + off; off += (size_t)BB * G4H;
    float* cath  = ws + off; off += (size_t)BB * H2;       // [w | hy]
    float* xg    = ws + off;
    const size_t need_xg = (off + (size_t)TT * BB * G4H) * sizeof(float);
    const int precomp = (ws_size >= need_xg) ? 1 : 0;

    // One-time weight transposes (K-major for WMMA B-fragments)
    {
        dim3 tb(32, 8);
        transpose_kernel<<<dim3(INW / 32, G4H / 32), tb, 0, stream>>>(Wi, WiT, G4H, INW);
        transpose_kernel<<<dim3(HH / 32, G4H / 32), tb, 0, stream>>>(Wh, WhT, G4H, HH);
        transpose_kernel<<<dim3(H2 / 32, HH / 32), tb, 0, stream>>>(W_out, WoT, HH, H2);
    }

    // ctxP = ctx @ W_in : 4096x512x512 -> 1024 blocks x 8 wave-tiles
    ctxp_kernel<<<(BB * SS / 16) * (HH / 16) / 8, 256, 0, stream>>>(ctx, W_in, ctxP);

    // h/c state init
    hipMemcpyAsync(hbuf, h0, (size_t)BB * HH * sizeof(float),
                   hipMemcpyDeviceToDevice, stream);
    hipMemcpyAsync(cbuf, c0, (size_t)BB * HH * sizeof(float),
                   hipMemcpyDeviceToDevice, stream);

    if (precomp) {
        // 8192x2048x512 GEMM -> 8192 blocks x 8 wave-tiles
        const int nblocks = (TT * BB / 16) * (G4H / 16) / 8;
        xgates_kernel<<<nblocks, 256, 0, stream>>>(x, WiT, bi, bh, xg);
    }

    for (int t = 0; t < TT; ++t) {
        const float* xg_t = precomp ? (xg + (size_t)t * BB * G4H) : nullptr;
        const float* x_t  = x + (size_t)t * INW;
        step_gates_kernel<<<32, 256, 0, stream>>>(hbuf, WhT, xg_t, x_t, WiT,
                                                  bi, bh, gates, precomp ? 0 : 1);
        step_attn_kernel<<<BB, 512, 0, stream>>>(gates, cbuf, ctx, ctxP, cmask, cath);
        step_htilde_kernel<<<8, 256, 0, stream>>>(cath, WoT, hbuf,
                                                  out + (size_t)t * HH);
    }

    // Final h_n, c_n appended after output [B,T,H]
    hipMemcpyAsync(out + (size_t)BB * TT * HH, hbuf,
                   (size_t)BB * HH * sizeof(float), hipMemcpyDeviceToDevice, stream);
    hipMemcpyAsync(out + (size_t)BB * TT * HH + (size_t)BB * HH, cbuf,
                   (size_t)BB * HH * sizeof(float), hipMemcpyDeviceToDevice, stream);
}